// SparseConvolutionDownsample_36550171689401
// MI455X (gfx1250) — compile-verified
//
#include <hip/hip_runtime.h>
#include <hip/hip_bf16.h>

typedef __attribute__((ext_vector_type(2))) float v2f;
typedef __attribute__((ext_vector_type(8))) float v8f;
typedef __attribute__((ext_vector_type(2))) int   v2i;

#define C_IN   128
#define C_OUT  256
#define NTAPS  4
#define BN_EPS 1e-4f
#define WAVES_PER_BLOCK 32
#define GEMM_BLOCK (WAVES_PER_BLOCK * 32)
#define BFRAG_PAIRS 512                 /* 32 K-steps x 16 N-tiles */
#define BFRAG_BYTES (BFRAG_PAIRS * 32 * 8) /* 128 KB */

#define AS_GLOBAL __attribute__((address_space(1)))
#define AS_LDS    __attribute__((address_space(3)))

#if defined(__gfx1250__) && \
    __has_builtin(__builtin_amdgcn_global_load_async_to_lds_b64) && \
    __has_builtin(__builtin_amdgcn_s_wait_asynccnt)
#define USE_ASYNC_LDS 1
#else
#define USE_ASYNC_LDS 0
#endif

// ---------------------------------------------------------------------------
// 1) Transpose weights W[k][K][n] -> Wt[k][n][K] (contiguous K pairs).
// ---------------------------------------------------------------------------
__global__ void wt_transpose_kernel(const float* __restrict__ W,
                                    float* __restrict__ Wt) {
    int idx = blockIdx.x * blockDim.x + threadIdx.x;
    if (idx >= NTAPS * C_IN * C_OUT) return;
    int k = idx / (C_IN * C_OUT);
    int r = idx - k * (C_IN * C_OUT);
    int K = r / C_OUT;
    int n = r - K * C_OUT;
    Wt[((long)(k * C_OUT + n)) * C_IN + K] = W[idx];
}

// ---------------------------------------------------------------------------
// 2) Counting-sort rows by kernel tap: perm[k*N + rank] = row index.
// ---------------------------------------------------------------------------
__global__ void tap_scatter_kernel(const int* __restrict__ offset,
                                   int* __restrict__ perm,
                                   int* __restrict__ tapPtr, int N) {
    int i = blockIdx.x * blockDim.x + threadIdx.x;
    if (i >= N) return;
    int k = offset[i];
    int pos = atomicAdd(&tapPtr[k], 1);
    perm[(long)k * N + pos] = i;
}

// ---------------------------------------------------------------------------
// 3) WMMA fp32 GEMM + scatter.
//    Block = 32 waves, all on the same tap k.  The block cooperatively stages
//    the tap's 256x128 weight panel into LDS *in B-fragment order* via async
//    global->LDS (per-lane scatter addresses), so the hot loop's B loads are
//    conflict-free ds_load_b64.  Each wave then owns one 16-row tile: 16
//    N-tiles x 32 chained v_wmma_f32_16x16x4_f32, scatter-added to yconv.
// ---------------------------------------------------------------------------
__global__ void __launch_bounds__(GEMM_BLOCK)
sparse_gemm_scatter_kernel(const float* __restrict__ x,
                           const float* __restrict__ Wt,
                           const int* __restrict__ perm,
                           const int* __restrict__ tapPtr,
                           const int* __restrict__ out_index,
                           float* __restrict__ yconv,
                           int N, int blocksPerTap) {
    extern __shared__ v2f bfrag[];               // [BFRAG_PAIRS][32], 128 KB
    const int lane = threadIdx.x & 31;
    const int wave = threadIdx.x >> 5;
    const int k  = blockIdx.x / blocksPerTap;    // block-uniform tap
    const int tb = blockIdx.x - k * blocksPerTap;
    const int cnt = tapPtr[k];                   // rows in this tap segment
    if (tb * (WAVES_PER_BLOCK * 16) >= cnt) return;  // whole block idle

    const int m     = lane & 15;                 // row / channel slot of lane
    const int khalf = (lane >> 4) * 2;           // lanes 0-15: K 0,1; 16-31: K 2,3

    // ---- cooperative B-panel stage: fragment-ordered gather into LDS ----
    const float* wb = Wt + (long)k * C_OUT * C_IN;
    for (int p = wave; p < BFRAG_PAIRS; p += WAVES_PER_BLOCK) {
        int kk = p >> 4;
        int nt = p & 15;
        int n  = nt * 16 + m;
        const float* src = wb + (long)n * C_IN + kk * 4 + khalf;
#if USE_ASYNC_LDS
        __builtin_amdgcn_global_load_async_to_lds_b64(
            (AS_GLOBAL v2i*)src,
            (AS_LDS v2i*)&bfrag[p * 32 + lane],
            0, 0);
#else
        bfrag[p * 32 + lane] = *(const v2f*)src;
#endif
    }
#if USE_ASYNC_LDS
    __builtin_amdgcn_s_wait_asynccnt(0);
#endif
    __syncthreads();

    const int t    = tb * WAVES_PER_BLOCK + wave;    // this wave's tile
    const int row0 = t * 16;
    if (row0 >= cnt) return;                         // after barrier: safe

    const bool valid = (row0 + m) < cnt;
    int srcRow = valid ? perm[(long)k * N + row0 + m] : 0;
    int outRow = valid ? out_index[srcRow] : 0;      // invalid rows add 0 to row 0

    // A fragments for the whole K=128 extent: 32 steps of 16x4.
    v2f a[32];
    const float* xrow = x + (long)srcRow * C_IN + khalf;
#pragma unroll
    for (int kk = 0; kk < 32; ++kk) {
        v2f av = {0.0f, 0.0f};
        if (valid) av = *(const v2f*)(xrow + kk * 4);
        a[kk] = av;
    }

    for (int nt = 0; nt < 16; ++nt) {
        v8f acc = {};
#pragma unroll
        for (int kk = 0; kk < 32; ++kk) {
            v2f b = bfrag[(kk * 16 + nt) * 32 + lane];
            acc = __builtin_amdgcn_wmma_f32_16x16x4_f32(
                false, a[kk], false, b, (short)0, acc, false, false);
        }
        // D layout: VGPR j holds M=j (lanes 0-15) / M=j+8 (lanes 16-31), N=lane&15.
        int nc = nt * 16 + m;
#pragma unroll
        for (int j = 0; j < 8; ++j) {
            int mrow = j + ((lane >> 4) << 3);
            int orow = __shfl(outRow, mrow, 32);
            unsafeAtomicAdd(yconv + (long)orow * C_OUT + nc, acc[j]);
        }
    }
}

// ---------------------------------------------------------------------------
// 4) Per-channel sum / sum-of-squares over yconv (padded rows are zero).
// ---------------------------------------------------------------------------
__global__ void channel_reduce_kernel(const float* __restrict__ yconv,
                                      float* __restrict__ gSum,
                                      float* __restrict__ gSumSq, int N) {
    int c = threadIdx.x;                         // 256 threads = 256 channels
    float s = 0.0f, sq = 0.0f;
    for (int row = blockIdx.x; row < N; row += gridDim.x) {
        float v = yconv[(long)row * C_OUT + c];
        s += v; sq += v * v;
    }
    unsafeAtomicAdd(&gSum[c], s);
    unsafeAtomicAdd(&gSumSq[c], sq);
}

// ---------------------------------------------------------------------------
// 5) cnt = sum(out_mask)
// ---------------------------------------------------------------------------
__global__ void mask_reduce_kernel(const float* __restrict__ mask,
                                   float* __restrict__ gCnt, int N) {
    int i0 = blockIdx.x * blockDim.x + threadIdx.x;
    int stride = gridDim.x * blockDim.x;
    float s = 0.0f;
    for (int i = i0; i < N; i += stride) s += mask[i];
    for (int off = 16; off > 0; off >>= 1) s += __shfl_down(s, off, 32);
    if ((threadIdx.x & 31) == 0) unsafeAtomicAdd(gCnt, s);
}

// ---------------------------------------------------------------------------
// 6) Fold BN stats into per-channel scale/bias.
// ---------------------------------------------------------------------------
__global__ void bn_params_kernel(const float* __restrict__ gSum,
                                 const float* __restrict__ gSumSq,
                                 const float* __restrict__ gCnt,
                                 const float* __restrict__ gamma,
                                 const float* __restrict__ beta,
                                 float* __restrict__ scale,
                                 float* __restrict__ bias) {
    int c = threadIdx.x;
    float cnt  = *gCnt;
    float mean = gSum[c] / cnt;
    float var  = gSumSq[c] / cnt - mean * mean;
    float sc   = gamma[c] * rsqrtf(var + BN_EPS);
    scale[c] = sc;
    bias[c]  = beta[c] - mean * sc;
}

// ---------------------------------------------------------------------------
// 7) In-place BN + ReLU + mask on yconv (= d_out).
// ---------------------------------------------------------------------------
__global__ void bn_relu_kernel(float* __restrict__ y,
                               const float* __restrict__ scale,
                               const float* __restrict__ bias,
                               const float* __restrict__ mask, long total) {
    long e = (long)blockIdx.x * blockDim.x + threadIdx.x;
    if (e >= total) return;
    int  c   = (int)(e & (C_OUT - 1));
    long row = e >> 8;                           // C_OUT == 256
    float v = y[e];
    float o = fmaxf(fmaf(v, scale[c], bias[c]), 0.0f) * mask[row];
    y[e] = o;
}

// ---------------------------------------------------------------------------
extern "C" void kernel_launch(void* const* d_in, const int* in_sizes, int n_in,
                              void* d_out, int out_size, void* d_ws, size_t ws_size,
                              hipStream_t stream) {
    const float* x        = (const float*)d_in[0];
    const float* W        = (const float*)d_in[1];
    const float* gamma    = (const float*)d_in[2];
    const float* beta     = (const float*)d_in[3];
    const int*   offset   = (const int*)d_in[4];
    const int*   out_idx  = (const int*)d_in[5];
    const float* out_mask = (const float*)d_in[6];
    float* yconv = (float*)d_out;
    const int N = in_sizes[0] / C_IN;

    // workspace layout
    char* ws = (char*)d_ws;
    float* Wt     = (float*)ws;  ws += (size_t)NTAPS * C_IN * C_OUT * sizeof(float);
    int*   perm   = (int*)ws;    ws += (size_t)NTAPS * N * sizeof(int);
    int*   tapPtr = (int*)ws;    ws += 4 * sizeof(int);
    float* gSum   = (float*)ws;  ws += C_OUT * sizeof(float);
    float* gSumSq = (float*)ws;  ws += C_OUT * sizeof(float);
    float* gCnt   = (float*)ws;  ws += sizeof(float);
    float* scale  = (float*)ws;  ws += C_OUT * sizeof(float);
    float* bias   = (float*)ws;  ws += C_OUT * sizeof(float);

    // zero accumulator (d_out) and counters/stats (tapPtr..gCnt contiguous)
    (void)hipMemsetAsync(d_out, 0, (size_t)N * C_OUT * sizeof(float), stream);
    (void)hipMemsetAsync(tapPtr, 0,
                         4 * sizeof(int) + (2 * C_OUT + 1) * sizeof(float), stream);

    wt_transpose_kernel<<<(NTAPS * C_IN * C_OUT + 255) / 256, 256, 0, stream>>>(W, Wt);
    tap_scatter_kernel<<<(N + 255) / 256, 256, 0, stream>>>(offset, perm, tapPtr, N);

    const int TPT = (N + 15) / 16;               // max 16-row tiles per tap
    const int blocksPerTap = (TPT + WAVES_PER_BLOCK - 1) / WAVES_PER_BLOCK;
    sparse_gemm_scatter_kernel<<<NTAPS * blocksPerTap, GEMM_BLOCK, BFRAG_BYTES, stream>>>(
        x, Wt, perm, tapPtr, out_idx, yconv, N, blocksPerTap);

    channel_reduce_kernel<<<2048, 256, 0, stream>>>(yconv, gSum, gSumSq, N);
    mask_reduce_kernel<<<256, 256, 0, stream>>>(out_mask, gCnt, N);
    bn_params_kernel<<<1, C_OUT, 0, stream>>>(gSum, gSumSq, gCnt, gamma, beta, scale, bias);

    const long total = (long)N * C_OUT;
    bn_relu_kernel<<<(int)((total + 255) / 256), 256, 0, stream>>>(
        yconv, scale, bias, out_mask, total);
}